// Model_66125316489906
// MI455X (gfx1250) — compile-verified
//
#include <hip/hip_runtime.h>

#define N_NODES 50000
#define N_EDGES 600000
#define N_SUP   200000
#define NH      (N_NODES * 128)   // node x 128 feature plane

typedef float v2f __attribute__((ext_vector_type(2)));
typedef float v8f __attribute__((ext_vector_type(8)));

// ---------------------------------------------------------------------------
// GEMM: out[n,128] = diag(rowscale?) * [A0|A1|A2][n,K] @ W[K,128] (+ bias?)
// One block: 64 rows x 128 cols. 8 waves; wave w owns 16-col tile w and all
// four 16-row tiles (4 accumulators). K in {128,384}, staged in 64-wide
// K chunks in LDS. 64 x V_WMMA_F32_16X16X4_F32 per chunk per wave.
// ---------------------------------------------------------------------------
__global__ __launch_bounds__(256) void k_gemm(
    const float* __restrict__ A0, const float* __restrict__ A1,
    const float* __restrict__ A2, const float* __restrict__ rowscale,
    const float* __restrict__ W,  const float* __restrict__ bias,
    float* __restrict__ out, int K, int n)
{
    __shared__ float As[64 * 68];    // 64 rows x 64 k  (+4 pad: conflict-free)
    __shared__ float Wt[128 * 70];   // 128 cols x 64 k (+6 pad, even stride)

    const int tid  = threadIdx.x;
    const int wave = tid >> 5;
    const int lane = tid & 31;
    const int lo   = lane & 15;      // row/col within 16x16 tile
    const int hi   = lane >> 4;      // K sub-pair selector (f32 16x16x4 layout)
    const int row0 = blockIdx.x * 64;
    const int col0 = wave * 16;

    const float* Asrc[3] = {A0, A1, A2};

    v8f acc[4] = {};
    const int nchunks = K >> 6;
    for (int c = 0; c < nchunks; ++c) {
        const int kg0 = c << 6;
        // Stage A chunk (row-scaled, row index clamped for the tail block)
        for (int t = tid; t < 64 * 64; t += 256) {
            const int r = t >> 6, k = t & 63;
            const int gk = kg0 + k;
            const int rr = min(row0 + r, n - 1);
            float v = Asrc[gk >> 7][(size_t)rr * 128 + (gk & 127)];
            if (rowscale) v *= rowscale[rr];
            As[r * 68 + k] = v;
        }
        // Stage W chunk transposed: Wt[n][k] = W[kg0+k][n]
        for (int t = tid; t < 64 * 128; t += 256) {
            const int k = t >> 7, nn = t & 127;
            Wt[nn * 70 + k] = W[(size_t)(kg0 + k) * 128 + nn];
        }
        __syncthreads();
        #pragma unroll
        for (int s = 0; s < 16; ++s) {
            const int kb = s * 4 + hi * 2;
            const v2f b = *(const v2f*)&Wt[(col0 + lo) * 70 + kb]; // W[kb..kb+1][col]
            #pragma unroll
            for (int rt = 0; rt < 4; ++rt) {
                const v2f a = *(const v2f*)&As[(rt * 16 + lo) * 68 + kb];
                acc[rt] = __builtin_amdgcn_wmma_f32_16x16x4_f32(
                    false, a, false, b, (short)0, acc[rt], false, false);
            }
        }
        __syncthreads();
    }
    const int cidx = col0 + lo;
    const float bv = bias ? bias[cidx] : 0.0f;
    #pragma unroll
    for (int rt = 0; rt < 4; ++rt) {
        #pragma unroll
        for (int v = 0; v < 8; ++v) {
            const int r = row0 + rt * 16 + v + hi * 8;   // C/D: VGPR v -> M=v / v+8
            if (r < n) out[(size_t)r * 128 + cidx] = acc[rt][v] + bv;
        }
    }
}

// ---------------------------------------------------------------------------
// Degree accumulation (f32 hardware atomics)
// ---------------------------------------------------------------------------
__global__ __launch_bounds__(256) void k_degrees(
    const int* __restrict__ ei, float* __restrict__ dout,
    float* __restrict__ din, int E)
{
    const int e = blockIdx.x * blockDim.x + threadIdx.x;
    if (e >= E) return;
    __hip_atomic_fetch_add(&dout[ei[e]],    1.0f, __ATOMIC_RELAXED, __HIP_MEMORY_SCOPE_AGENT);
    __hip_atomic_fetch_add(&din[ei[e + E]], 1.0f, __ATOMIC_RELAXED, __HIP_MEMORY_SCOPE_AGENT);
}

__global__ __launch_bounds__(256) void k_invsqrt(
    const float* __restrict__ dout, const float* __restrict__ din,
    float* __restrict__ rns, float* __restrict__ rnd, int n)
{
    const int i = blockIdx.x * blockDim.x + threadIdx.x;
    if (i >= n) return;
    rns[i] = 1.0f / sqrtf(fmaxf(dout[i], 1.0f));
    rnd[i] = 1.0f / sqrtf(fmaxf(din[i],  1.0f));
}

// ---------------------------------------------------------------------------
// agg[dst] += x[src] ; one thread moves 4 contiguous floats of one edge
// ---------------------------------------------------------------------------
__global__ __launch_bounds__(256) void k_scatter(
    const float* __restrict__ x, const int* __restrict__ ei,
    float* __restrict__ agg, int E)
{
    const int gid = blockIdx.x * blockDim.x + threadIdx.x;
    if (gid >= E * 32) return;
    const int e = gid >> 5;
    const int q = (gid & 31) << 2;
    const int s = ei[e];
    const int d = ei[e + E];
    const float4 v = *(const float4*)(x + (size_t)s * 128 + q);
    float* p = agg + (size_t)d * 128 + q;
    __hip_atomic_fetch_add(p + 0, v.x, __ATOMIC_RELAXED, __HIP_MEMORY_SCOPE_AGENT);
    __hip_atomic_fetch_add(p + 1, v.y, __ATOMIC_RELAXED, __HIP_MEMORY_SCOPE_AGENT);
    __hip_atomic_fetch_add(p + 2, v.z, __ATOMIC_RELAXED, __HIP_MEMORY_SCOPE_AGENT);
    __hip_atomic_fetch_add(p + 3, v.w, __ATOMIC_RELAXED, __HIP_MEMORY_SCOPE_AGENT);
}

// h = relu(agg * norm_dst + b)   (in place)
__global__ __launch_bounds__(256) void k_postagg(
    float* __restrict__ h, const float* __restrict__ rnd,
    const float* __restrict__ b, int n)
{
    const int i = blockIdx.x * blockDim.x + threadIdx.x;
    if (i >= n) return;
    h[i] = fmaxf(h[i] * rnd[i >> 7] + b[i & 127], 0.0f);
}

// ---------------------------------------------------------------------------
// Cosine scoring: one wave32 per support edge (pos then neg, contiguous out)
// ---------------------------------------------------------------------------
__global__ __launch_bounds__(256) void k_predict(
    const int* __restrict__ ep, const int* __restrict__ en,
    const float* __restrict__ hf, const float* __restrict__ hs,
    float* __restrict__ out, int nsup)
{
    const int gw = (blockIdx.x * blockDim.x + threadIdx.x) >> 5;
    const int lane = threadIdx.x & 31;
    if (gw >= 2 * nsup) return;
    const int* e = (gw < nsup) ? ep : en;
    const int idx = (gw < nsup) ? gw : gw - nsup;
    const int s = e[idx];
    const int d = e[idx + nsup];
    const float* a = hf + (size_t)s * 128;
    const float* c = hs + (size_t)d * 128;
    float aa = 0.f, cc = 0.f, ac = 0.f;
    #pragma unroll
    for (int j = 0; j < 4; ++j) {
        const float av = a[lane + 32 * j];
        const float cv = c[lane + 32 * j];
        aa += av * av; cc += cv * cv; ac += av * cv;
    }
    #pragma unroll
    for (int off = 16; off > 0; off >>= 1) {
        aa += __shfl_down(aa, off, 32);
        cc += __shfl_down(cc, off, 32);
        ac += __shfl_down(ac, off, 32);
    }
    if (lane == 0) out[gw] = ac / sqrtf(aa * cc);
}

// ---------------------------------------------------------------------------
extern "C" void kernel_launch(void* const* d_in, const int* in_sizes, int n_in,
                              void* d_out, int out_size, void* d_ws, size_t ws_size,
                              hipStream_t stream) {
    const float* feat1 = (const float*)d_in[0];
    const float* feat2 = (const float*)d_in[1];
    const int*   ei1   = (const int*)d_in[2];
    const int*   ei2   = (const int*)d_in[3];
    const int*   ep    = (const int*)d_in[4];
    const int*   en    = (const int*)d_in[5];
    const float* w0    = (const float*)d_in[6];
    const float* b0    = (const float*)d_in[7];
    const float* w1    = (const float*)d_in[8];
    const float* b1    = (const float*)d_in[9];
    const float* wfst  = (const float*)d_in[10];
    const float* bfst  = (const float*)d_in[11];
    const float* wsnd  = (const float*)d_in[12];
    const float* bsnd  = (const float*)d_in[13];

    float* out     = (float*)d_out;
    float* posneg  = out;                        // [2*N_SUP]
    float* hfirst  = out + 2 * N_SUP;            // [N_NODES,128]
    float* hsecond = hfirst + NH;                // [N_NODES,128]

    float* fws  = (float*)d_ws;
    float* bufX = fws;                // N x 128 (x = scaled A @ W)
    float* bufB = bufX + NH;          // N x 128 (agg1 -> h1)
    float* bufC = bufB + NH;          // N x 128 (agg2 -> h2)
    float* degO = bufC + NH;          // N
    float* degI = degO + N_NODES;     // N
    float* rns  = degI + N_NODES;     // N
    float* rnd  = rns  + N_NODES;     // N

    const int gemmBlocks = (N_NODES + 63) / 64;  // 782 (last block row-clamped)
    const int nThread    = (NH + 255) / 256;
    const int eThread    = (N_EDGES * 32 + 255) / 256;

    for (int g = 0; g < 2; ++g) {
        const float* feat = g ? feat2 : feat1;
        const int*   ei   = g ? ei2   : ei1;
        const float* wF   = g ? wsnd  : wfst;
        const float* bF   = g ? bsnd  : bfst;
        float*       hout = g ? hsecond : hfirst;

        // degrees + symmetric norms
        hipMemsetAsync(degO, 0, 2 * N_NODES * sizeof(float), stream);
        k_degrees<<<(N_EDGES + 255) / 256, 256, 0, stream>>>(ei, degO, degI, N_EDGES);
        k_invsqrt<<<(N_NODES + 255) / 256, 256, 0, stream>>>(degO, degI, rns, rnd, N_NODES);

        // layer 0: x = (feat * rns) @ w0 ; agg ; h1 = relu(agg*rnd + b0)
        k_gemm<<<gemmBlocks, 256, 0, stream>>>(feat, nullptr, nullptr, rns, w0, nullptr, bufX, 128, N_NODES);
        hipMemsetAsync(bufB, 0, (size_t)NH * sizeof(float), stream);
        k_scatter<<<eThread, 256, 0, stream>>>(bufX, ei, bufB, N_EDGES);
        k_postagg<<<nThread, 256, 0, stream>>>(bufB, rnd, b0, NH);

        // layer 1: x = (h1 * rns) @ w1 ; agg ; h2 = relu(agg*rnd + b1)
        k_gemm<<<gemmBlocks, 256, 0, stream>>>(bufB, nullptr, nullptr, rns, w1, nullptr, bufX, 128, N_NODES);
        hipMemsetAsync(bufC, 0, (size_t)NH * sizeof(float), stream);
        k_scatter<<<eThread, 256, 0, stream>>>(bufX, ei, bufC, N_EDGES);
        k_postagg<<<nThread, 256, 0, stream>>>(bufC, rnd, b1, NH);

        // head: hout = [feat|h1|h2] @ wF + bF   (K = 384)
        k_gemm<<<gemmBlocks, 256, 0, stream>>>(feat, bufB, bufC, nullptr, wF, bF, hout, 384, N_NODES);
    }

    // cosine similarity for pos + neg support edges
    k_predict<<<(2 * N_SUP * 32 + 255) / 256, 256, 0, stream>>>(ep, en, hfirst, hsecond, posneg, N_SUP);
}